// SSDLoss_31576599561062
// MI455X (gfx1250) — compile-verified
//
#include <hip/hip_runtime.h>
#include <stdint.h>

#define B_   32
#define T_   32
#define A_   24576
#define C_   80
#define IMG_ 224.0f

// ---- k_loss tiling: 96 rows/tile, double-buffered, 8 tiles per block ------
#define TILE_ROWS 96
#define ROW_F     81                        // C+1 floats per pred_label row
#define TILE_F    (TILE_ROWS * ROW_F)       // 7776 floats = 31104 B
#define TILE_V4   (TILE_F / 4)              // 1944 16-byte vectors
#define TILE_V4_BASE (TILE_V4 / TILE_ROWS)  // 20 full rounds
#define TILES_PER_BLOCK 8
#define BLOCKS_PER_SAMPLE (A_ / (TILE_ROWS * TILES_PER_BLOCK))   // 32

// -------- pack (iou, anchor) for first-index argmax via atomicMax ----------
__device__ __forceinline__ unsigned long long pack_iou(float iou, int a) {
    // iou >= 0, so float bits are order-preserving as unsigned.
    return (((unsigned long long)__float_as_uint(iou)) << 32) |
           (unsigned int)(A_ - 1 - a);
}

__device__ __forceinline__ void async_b128(unsigned int lds, unsigned int off,
                                           unsigned long long gbase) {
    asm volatile("global_load_async_to_lds_b128 %0, %1, %2"
                 :: "v"(lds), "v"(off), "s"(gbase) : "memory");
}

// Issue one tile (96 rows x 81 f32) into LDS. Every wave issues EXACTLY 21
// async b128 ops: 20 uniform rounds + 1 remainder op on 8 lanes per wave.
__device__ __forceinline__ void issue_tile(unsigned int ldsbase,
                                           unsigned long long gbase, int tid) {
    #pragma unroll
    for (int j = 0; j < TILE_V4_BASE; ++j) {
        unsigned int off = (unsigned int)(j * TILE_ROWS + tid) * 16u;
        async_b128(ldsbase + off, off, gbase);
    }
    if ((tid & 3) == 0) {   // 24 remainder vectors: 8 lanes in each of 3 waves
        unsigned int off = (unsigned int)(TILE_V4_BASE * TILE_ROWS + (tid >> 2)) * 16u;
        async_b128(ldsbase + off, off, gbase);
    }
}

// ---------------- K0: zero accumulators (must run every launch) ------------
__global__ void k_init(unsigned long long* gbest, float* bbsum, float* npos,
                       float* lblsum) {
    int i = blockIdx.x * blockDim.x + threadIdx.x;
    if (i < B_ * T_) gbest[i] = 0ull;
    if (i < B_) { bbsum[i] = 0.f; npos[i] = 0.f; lblsum[i] = 0.f; }
}

// ---------------- K1: pairwise IoU, per-anchor max/argmax, per-target best -
__global__ __launch_bounds__(256) void k_iou(
    const float* __restrict__ anchors, const float* __restrict__ tbb,
    float* __restrict__ aval, int* __restrict__ aidx,
    unsigned long long* __restrict__ gbest) {
    __shared__ float4 tb[T_];
    __shared__ float  tarea[T_];
    __shared__ unsigned long long sbest[T_];

    const int tid = threadIdx.x;
    const int b   = blockIdx.x / (A_ / 256);
    const int a   = (blockIdx.x % (A_ / 256)) * 256 + tid;

    if (tid < T_) {
        float4 t  = ((const float4*)tbb)[b * T_ + tid];
        tb[tid]   = t;
        tarea[tid]= (t.z - t.x) * (t.w - t.y);
        sbest[tid]= 0ull;
    }
    __syncthreads();

    const float4 an = ((const float4*)anchors)[a];   // (cx, cy, w, h)
    const float ax0 = an.x - 0.5f * an.z, ay0 = an.y - 0.5f * an.w;
    const float ax1 = an.x + 0.5f * an.z, ay1 = an.y + 0.5f * an.w;
    const float area_a = an.z * an.w;

    float best = -1.0f; int bi = 0;
    const int lane = tid & 31;

    for (int t = 0; t < T_; ++t) {
        float4 tc = tb[t];
        float iw = fmaxf(fminf(ax1, tc.z) - fmaxf(ax0, tc.x), 0.f);
        float ih = fmaxf(fminf(ay1, tc.w) - fmaxf(ay0, tc.y), 0.f);
        float inter = iw * ih;
        float iou   = inter / (area_a + tarea[t] - inter);
        if (iou > best) { best = iou; bi = t; }   // first-max wins (argmax)

        unsigned long long pk = pack_iou(iou, a);
        #pragma unroll
        for (int off = 16; off > 0; off >>= 1) {
            unsigned long long o = __shfl_down(pk, off, 32);
            if (o > pk) pk = o;
        }
        if (lane == 0) atomicMax(&sbest[t], pk);
    }

    const int g = b * A_ + a;
    aval[g] = best;
    aidx[g] = bi;

    __syncthreads();
    if (tid < T_) atomicMax(&gbest[b * T_ + tid], sbest[tid]);
}

// ---------------- K2: .at[prior_idx].set(...) — sequential t => last wins --
__global__ void k_override(const unsigned long long* __restrict__ gbest,
                           float* __restrict__ aval, int* __restrict__ aidx) {
    int b = threadIdx.x;
    if (b >= B_) return;
    for (int t = 0; t < T_; ++t) {
        unsigned long long pk = gbest[b * T_ + t];
        int a = A_ - 1 - (int)(unsigned int)(pk & 0xffffffffu);
        aval[b * A_ + a] = 1.99f;
        aidx[b * A_ + a] = t;
    }
}

// Negative-class focal term: 0.75 * p * (max(x,0) + log1p(exp(-|x|)))
// e = exp(-|x|) in (0,1] => 1+e in (1,2]: fast rcp/log are precision-safe.
__device__ __forceinline__ void focal_terms(float x, float& neg, float& p) {
    float e = __expf(-fabsf(x));
    float l = __logf(1.f + e);
    float r = __builtin_amdgcn_rcpf(1.f + e);
    p = (x >= 0.f) ? r : e * r;
    neg = 0.75f * p * (fmaxf(x, 0.f) + l);
}

// ---------------- K3: focal + bbox losses; double-buffered async-LDS stage -
__global__ __launch_bounds__(TILE_ROWS) void k_loss(
    const float* __restrict__ anchors, const float* __restrict__ tbb,
    const int*   __restrict__ tlbl,    const float* __restrict__ pbb,
    const float* __restrict__ plbl,    const float* __restrict__ aval,
    const int*   __restrict__ aidx,
    float* __restrict__ bbsum, float* __restrict__ npos,
    float* __restrict__ lblsum) {
    __shared__ __align__(16) float tile[2][TILE_F];   // 2 x 31104 B
    __shared__ float red[12];

    const int tid = threadIdx.x;
    const int b   = blockIdx.x / BLOCKS_PER_SAMPLE;
    const int firstRow = (blockIdx.x % BLOCKS_PER_SAMPLE) *
                         (TILES_PER_BLOCK * TILE_ROWS);

    const unsigned long long gbase0 =
        (unsigned long long)(uintptr_t)(plbl + ((size_t)b * A_ + firstRow) * ROW_F);
    const unsigned int lds0 = (unsigned int)(uintptr_t)(&tile[0][0]);
    const unsigned int lds1 = (unsigned int)(uintptr_t)(&tile[1][0]);
    const unsigned long long tileBytes = (unsigned long long)TILE_F * 4ull;

    // prologue: stage tile 0 into buffer 0 (21 async ops per wave)
    issue_tile(lds0, gbase0, tid);

    float lsum = 0.f, bsum = 0.f, np = 0.f;
    const float inv = 1.0f / IMG_;

    for (int t = 0; t < TILES_PER_BLOCK; ++t) {
        // issue next tile into the other buffer (it was consumed at t-1)
        if (t + 1 < TILES_PER_BLOCK) {
            unsigned int ldsn = ((t + 1) & 1) ? lds1 : lds0;
            issue_tile(ldsn, gbase0 + (unsigned long long)(t + 1) * tileBytes, tid);
            // 42 outstanding -> wait to 21 retires exactly tile t (in-order)
            asm volatile("s_wait_asynccnt 21" ::: "memory");
        } else {
            asm volatile("s_wait_asynccnt 0" ::: "memory");
        }
        __syncthreads();   // tile t fully visible to all waves

        const float* row = ((t & 1) ? &tile[1][0] : &tile[0][0]) + tid * ROW_F;
        const int a = firstRow + t * TILE_ROWS + tid;
        const int g = b * A_ + a;

        const float av  = aval[g];
        const int   idx = aidx[g];
        const bool  pos = av > 0.5f;
        const int   cls = pos ? tlbl[b * T_ + idx] : C_;  // BG => one_hot all 0

        // focal loss: generic negative form for all 80 cols, patch cls column
        float acc = 0.f;
        for (int c = 0; c < C_; ++c) {
            float negterm, p;
            focal_terms(row[c], negterm, p);
            acc += negterm;
        }
        if (cls < C_) {
            float x = row[cls];
            float negterm, p;
            focal_terms(x, negterm, p);
            float e = __expf(-fabsf(x));
            float l = __logf(1.f + e);
            float posterm = 0.25f * (1.f - p) * (fmaxf(x, 0.f) - x + l);
            acc += posterm - negterm;
        }
        lsum += acc;

        // bbox loss: tanh decode vs matched target (ratio space)
        float4 pb = ((const float4*)pbb)[g];
        float4 an = ((const float4*)anchors)[a];
        float acx = an.x * inv, acy = an.y * inv, aw = an.z * inv, ah = an.w * inv;
        float t0 = tanhf(pb.x), t1 = tanhf(pb.y);
        float t2 = tanhf(pb.z), t3 = tanhf(pb.w);
        float cx = acx + 0.5f * t0 * aw, cy = acy + 0.5f * t1 * ah;
        float w  = aw * (0.5f * t2 + 1.f), h = ah * (0.5f * t3 + 1.f);
        float px0 = cx - 0.5f * w, py0 = cy - 0.5f * h;
        float px1 = cx + 0.5f * w, py1 = cy + 0.5f * h;
        float4 m = ((const float4*)tbb)[b * T_ + idx];
        if (pos) {
            bsum += fabsf(px0 - m.x * inv) + fabsf(py0 - m.y * inv) +
                    fabsf(px1 - m.z * inv) + fabsf(py1 - m.w * inv);
            np += 1.f;
        }
        __syncthreads();   // all waves done with tile t before its buffer reuse
    }

    // ---- block reduce (wave32 shuffles, then 3 wave partials)
    #pragma unroll
    for (int off = 16; off > 0; off >>= 1) {
        lsum += __shfl_down(lsum, off, 32);
        bsum += __shfl_down(bsum, off, 32);
        np   += __shfl_down(np,   off, 32);
    }
    const int wv = tid >> 5, lane = tid & 31;
    if (lane == 0) { red[wv] = lsum; red[4 + wv] = bsum; red[8 + wv] = np; }
    __syncthreads();
    if (tid == 0) {
        float L = 0.f, Bs = 0.f, N = 0.f;
        #pragma unroll
        for (int i = 0; i < TILE_ROWS / 32; ++i) {
            L += red[i]; Bs += red[4 + i]; N += red[8 + i];
        }
        atomicAdd(&lblsum[b], L);
        atomicAdd(&bbsum[b], Bs);
        atomicAdd(&npos[b], N);
    }
}

// ---------------- K4: per-sample normalize + final 2-scalar output ---------
__global__ void k_final(const float* __restrict__ bbsum,
                        const float* __restrict__ npos,
                        const float* __restrict__ lblsum,
                        float* __restrict__ out) {
    int b = threadIdx.x;
    float vb = 0.f, vl = 0.f;
    if (b < B_) {
        vb = bbsum[b] / fmaxf(npos[b] * 4.f, 1.f);
        vl = lblsum[b] * (1.0f / (float)C_);
    }
    #pragma unroll
    for (int off = 16; off > 0; off >>= 1) {
        vb += __shfl_down(vb, off, 32);
        vl += __shfl_down(vl, off, 32);
    }
    if (b == 0) { out[0] = vb; out[1] = vl; }
}

extern "C" void kernel_launch(void* const* d_in, const int* in_sizes, int n_in,
                              void* d_out, int out_size, void* d_ws,
                              size_t ws_size, hipStream_t stream) {
    const float* anchors = (const float*)d_in[0];   // (A,4) center form
    const float* tbb     = (const float*)d_in[1];   // (B,T,4) corners
    const int*   tlbl    = (const int*)  d_in[2];   // (B,T)
    const float* pbb     = (const float*)d_in[3];   // (B,A,4)
    const float* plbl    = (const float*)d_in[4];   // (B,A,81)

    char* ws = (char*)d_ws;
    float*              aval   = (float*)ws;                               // B*A f32
    int*                aidx   = (int*)(ws + 4ull * B_ * A_);              // B*A i32
    unsigned long long* gbest  = (unsigned long long*)(ws + 8ull * B_ * A_);
    float*              bbsum  = (float*)(ws + 8ull * B_ * A_ + 8ull * B_ * T_);
    float*              npos   = bbsum + B_;
    float*              lblsum = npos + B_;
    float*              out    = (float*)d_out;

    k_init<<<4, 256, 0, stream>>>(gbest, bbsum, npos, lblsum);
    k_iou<<<B_ * (A_ / 256), 256, 0, stream>>>(anchors, tbb, aval, aidx, gbest);
    k_override<<<1, 32, 0, stream>>>(gbest, aval, aidx);
    k_loss<<<B_ * BLOCKS_PER_SAMPLE, TILE_ROWS, 0, stream>>>(
        anchors, tbb, tlbl, pbb, plbl, aval, aidx, bbsum, npos, lblsum);
    k_final<<<1, 32, 0, stream>>>(bbsum, npos, lblsum, out);
}